// GATLayer_38663295599339
// MI455X (gfx1250) — compile-verified
//
#include <hip/hip_runtime.h>
#include <hip/hip_bf16.h>

typedef __attribute__((ext_vector_type(2))) float v2f;
typedef __attribute__((ext_vector_type(8))) float v8f;

#define GAT_N 100000
#define GAT_H 4
#define GAT_D 64
#define GAT_R 2
#define GAT_E 500000
#define GAT_HD 256
#define GAT_IN 256
#define TILE_ROWS 64
#define KC 32
#define WCOLS 272   // 256 cols + pad so row stride (544 floats) == 32 mod 64 banks

// ---------------- monotone float<->int encoding for atomic segment-max ----------------
__device__ __forceinline__ int enc_f32(float f) {
    int i = __float_as_int(f);
    return (i >= 0) ? i : (i ^ 0x7fffffff);
}
__device__ __forceinline__ float dec_f32(int k) {
    return __int_as_float((k >= 0) ? k : (k ^ 0x7fffffff));
}

// ---------------- out = bias_r0 + bias_r1 broadcast over nodes ----------------
__global__ __launch_bounds__(256) void out_init_kernel(const float* __restrict__ bias,
                                                       float* __restrict__ out, int total) {
    int i = blockIdx.x * 256 + threadIdx.x;
    if (i < total) {
        int c = i & (GAT_HD - 1);
        out[i] = bias[c] + bias[GAT_HD + c];
    }
}

// ---------------- per-relation segment state init ----------------
__global__ __launch_bounds__(256) void seg_init_kernel(int* __restrict__ mkey,
                                                       float* __restrict__ denom, int total) {
    int i = blockIdx.x * 256 + threadIdx.x;
    if (i < total) {
        mkey[i] = INT_MIN;   // decodes below any finite logit
        denom[i] = 0.0f;
    }
}

// ---------------- WMMA f32 GEMM: F[N,256] = X[N,256] @ W[256,256] ----------------
// Block: 256 threads (8 waves). Block tile: 64 rows x 256 cols.
// Wave w: row-tile w/2 (16 rows), col half w%2 (128 cols = 8 x 16x16 WMMA tiles).
// W staged in LDS pair-interleaved so each lane's B fragment is one aligned b64 load.
__global__ __launch_bounds__(256) void proj_gemm_kernel(const float* __restrict__ X,
                                                        const float* __restrict__ W,
                                                        float* __restrict__ F, int nrows) {
    __shared__ float xs[TILE_ROWS][KC + 2];       // 64 x 34   (~8.7 KB)
    __shared__ float wsm2[KC / 2][WCOLS][2];      // 16 x 272 x 2 (~34.8 KB)

    const int tid  = threadIdx.x;
    const int lane = tid & 31;
    const int wave = tid >> 5;
    const int rowBase = blockIdx.x * TILE_ROWS;

    const int rt = wave >> 1;   // row tile 0..3
    const int ch = wave & 1;    // col half 0..1

    v8f acc[8];
#pragma unroll
    for (int i = 0; i < 8; i++)
#pragma unroll
        for (int j = 0; j < 8; j++) acc[i][j] = 0.0f;

    for (int kc = 0; kc < GAT_IN; kc += KC) {
        // stage X tile: 32 rows x 32 cols, xs rows 32-63 left unwritten (clamped rows keep EXEC full)
        {
            int r = tid >> 3;
            int c = (tid & 7) << 2;
            int gr = rowBase + r;
            if (gr >= nrows) gr = nrows - 1;
            const float4 v = *reinterpret_cast<const float4*>(X + (size_t)gr * GAT_IN + kc + c);
            xs[r][c + 0] = v.x; xs[r][c + 1] = v.y; xs[r][c + 2] = v.z; xs[r][c + 3] = v.w;
        }
        // stage W tile pair-interleaved: wsm2[p][c] = { W[kc+2p][c], W[kc+2p+1][c] }
#pragma unroll
        for (int i = 0; i < 4; i++) {
            int p  = i * 4 + (tid >> 6);            // K pair row 0..15
            int cc = (tid & 63) << 2;
            const float4 v0 = *reinterpret_cast<const float4*>(W + (size_t)(kc + 2 * p)     * GAT_HD + cc);
            const float4 v1 = *reinterpret_cast<const float4*>(W + (size_t)(kc + 2 * p + 1) * GAT_HD + cc);
            wsm2[p][cc + 0][0] = v0.x; wsm2[p][cc + 0][1] = v1.x;
            wsm2[p][cc + 1][0] = v0.y; wsm2[p][cc + 1][1] = v1.y;
            wsm2[p][cc + 2][0] = v0.z; wsm2[p][cc + 2][1] = v1.z;
            wsm2[p][cc + 3][0] = v0.w; wsm2[p][cc + 3][1] = v1.w;
        }
        // prefetch next X chunk into cache hierarchy
        if (kc + KC < GAT_IN) {
            int r = tid >> 3;
            int gr = rowBase + r;
            if (gr >= nrows) gr = nrows - 1;
            __builtin_prefetch(X + (size_t)gr * GAT_IN + kc + KC, 0, 0);
        }
        __syncthreads();

        for (int k4 = 0; k4 < KC; k4 += 4) {
            // A fragment 16x4: lanes 0-15 -> K = k4+0,k4+1 ; lanes 16-31 -> K = k4+2,k4+3
            const int m  = lane & 15;
            const int kk = k4 + ((lane >> 4) << 1);
            const v2f a = *reinterpret_cast<const v2f*>(&xs[rt * 16 + m][kk]);

            // B fragment 4x16: one aligned b64 LDS load per lane, no packing moves
            const int kpair = (k4 >> 1) + (lane >> 4);
#pragma unroll
            for (int ct = 0; ct < 8; ct++) {
                const int c = ch * 128 + ct * 16 + (lane & 15);
                const v2f b = *reinterpret_cast<const v2f*>(&wsm2[kpair][c][0]);
                acc[ct] = __builtin_amdgcn_wmma_f32_16x16x4_f32(
                    false, a, false, b, (short)0, acc[ct], false, false);
            }
        }
        __syncthreads();
    }

    // C/D layout: VGPR v -> M = v (lanes 0-15) / v+8 (lanes 16-31); N = lane%16
#pragma unroll
    for (int ct = 0; ct < 8; ct++) {
        int col  = ch * 128 + ct * 16 + (lane & 15);
        int radd = (lane >> 4) << 3;
#pragma unroll
        for (int v = 0; v < 8; v++) {
            int gr = rowBase + rt * 16 + v + radd;
            if (gr < nrows) F[(size_t)gr * GAT_HD + col] = acc[ct][v];
        }
    }
}

// ---------------- el[n,h] = sum_d fs[n,h,d]*al[h,d] ; er likewise from fd ----------------
__global__ __launch_bounds__(256) void attn_dots_kernel(const float* __restrict__ fs,
                                                        const float* __restrict__ fd,
                                                        const float* __restrict__ al,
                                                        const float* __restrict__ ar,
                                                        float* __restrict__ el,
                                                        float* __restrict__ er) {
    __shared__ float s0[256];
    __shared__ float s1[256];
    const int node = blockIdx.x;
    const int t = threadIdx.x;
    s0[t] = fs[(size_t)node * GAT_HD + t] * al[t];
    s1[t] = fd[(size_t)node * GAT_HD + t] * ar[t];
    __syncthreads();
    for (int off = 32; off >= 1; off >>= 1) {
        if ((t & 63) < off) {
            s0[t] += s0[t + off];
            s1[t] += s1[t + off];
        }
        __syncthreads();
    }
    if ((t & 63) == 0) {
        int h = t >> 6;
        el[(size_t)node * GAT_H + h] = s0[t];
        er[(size_t)node * GAT_H + h] = s1[t];
    }
}

// ---------------- edge logits + segment max ----------------
__global__ __launch_bounds__(256) void edge_logits_kernel(const int* __restrict__ src,
                                                          const int* __restrict__ dst,
                                                          const float* __restrict__ el,
                                                          const float* __restrict__ er,
                                                          float* __restrict__ eatt,
                                                          int* __restrict__ mkey, int ne) {
    int e = blockIdx.x * 256 + threadIdx.x;
    if (e >= ne) return;
    int s = src[e], d = dst[e];
    float4 l = *reinterpret_cast<const float4*>(el + (size_t)s * GAT_H);
    float4 r = *reinterpret_cast<const float4*>(er + (size_t)d * GAT_H);
    float v[4] = { l.x + r.x, l.y + r.y, l.z + r.z, l.w + r.w };
#pragma unroll
    for (int h = 0; h < GAT_H; h++) {
        float x = v[h];
        x = (x > 0.0f) ? x : 0.2f * x;     // leaky_relu, slope 0.2
        eatt[(size_t)e * GAT_H + h] = x;
        atomicMax(mkey + (size_t)d * GAT_H + h, enc_f32(x));
    }
}

// ---------------- ee = exp(e - m[dst]) in place, denom = segment_sum(ee) ----------------
__global__ __launch_bounds__(256) void edge_exp_kernel(const int* __restrict__ dst,
                                                       float* __restrict__ eatt,
                                                       const int* __restrict__ mkey,
                                                       float* __restrict__ denom, int ne) {
    int e = blockIdx.x * 256 + threadIdx.x;
    if (e >= ne) return;
    int d = dst[e];
#pragma unroll
    for (int h = 0; h < GAT_H; h++) {
        float m  = dec_f32(mkey[(size_t)d * GAT_H + h]);
        float ee = __expf(eatt[(size_t)e * GAT_H + h] - m);
        eatt[(size_t)e * GAT_H + h] = ee;
        atomicAdd(denom + (size_t)d * GAT_H + h, ee);
    }
}

// ---------------- out[dst] += fs[src] * alpha ; 4 edges per 256-thread block ----------------
__global__ __launch_bounds__(256) void edge_msg_kernel(const int* __restrict__ src,
                                                       const int* __restrict__ dst,
                                                       const float* __restrict__ eatt,
                                                       const float* __restrict__ denom,
                                                       const float* __restrict__ fs,
                                                       float* __restrict__ out, int ne) {
    int e = blockIdx.x * 4 + (threadIdx.x >> 6);
    if (e >= ne) return;
    int t = threadIdx.x & 63;          // covers 256 cols as float4
    int s = src[e], d = dst[e];
    int h = t >> 4;                    // (t*4)/64
    float alpha = eatt[(size_t)e * GAT_H + h] / denom[(size_t)d * GAT_H + h];
    float4 f = *reinterpret_cast<const float4*>(fs + (size_t)s * GAT_HD + t * 4);
    float* o = out + (size_t)d * GAT_HD + t * 4;
    atomicAdd(o + 0, f.x * alpha);
    atomicAdd(o + 1, f.y * alpha);
    atomicAdd(o + 2, f.z * alpha);
    atomicAdd(o + 3, f.w * alpha);
}

extern "C" void kernel_launch(void* const* d_in, const int* in_sizes, int n_in,
                              void* d_out, int out_size, void* d_ws, size_t ws_size,
                              hipStream_t stream) {
    const float* x    = (const float*)d_in[0];
    const float* Wsrc = (const float*)d_in[1];
    const float* Wdst = (const float*)d_in[2];
    const float* al   = (const float*)d_in[3];
    const float* ar   = (const float*)d_in[4];
    const float* bias = (const float*)d_in[5];
    const int*   src  = (const int*)d_in[6];
    const int*   dst  = (const int*)d_in[7];
    float* out = (float*)d_out;

    // workspace layout
    float* fs    = (float*)d_ws;                       // N*256
    float* fd    = fs + (size_t)GAT_N * GAT_HD;        // N*256
    float* el    = fd + (size_t)GAT_N * GAT_HD;        // N*4
    float* er    = el + (size_t)GAT_N * GAT_H;         // N*4
    int*   mkey  = (int*)(er + (size_t)GAT_N * GAT_H); // N*4
    float* denom = (float*)(mkey + (size_t)GAT_N * GAT_H); // N*4
    float* eatt  = denom + (size_t)GAT_N * GAT_H;      // E*4

    out_init_kernel<<<(GAT_N * GAT_HD + 255) / 256, 256, 0, stream>>>(bias, out, GAT_N * GAT_HD);

    const int gemmGrid = (GAT_N + TILE_ROWS - 1) / TILE_ROWS;
    for (int r = 0; r < GAT_R; r++) {
        const float* wsp = Wsrc + (size_t)r * GAT_IN * GAT_HD;
        const float* wdp = Wdst + (size_t)r * GAT_IN * GAT_HD;
        const int* sp = src + (size_t)r * GAT_E;
        const int* dp = dst + (size_t)r * GAT_E;

        proj_gemm_kernel<<<gemmGrid, 256, 0, stream>>>(x, wsp, fs, GAT_N);
        proj_gemm_kernel<<<gemmGrid, 256, 0, stream>>>(x, wdp, fd, GAT_N);
        attn_dots_kernel<<<GAT_N, 256, 0, stream>>>(fs, fd, al + (size_t)r * GAT_HD,
                                                    ar + (size_t)r * GAT_HD, el, er);
        seg_init_kernel<<<(GAT_N * GAT_H + 255) / 256, 256, 0, stream>>>(mkey, denom, GAT_N * GAT_H);
        edge_logits_kernel<<<(GAT_E + 255) / 256, 256, 0, stream>>>(sp, dp, el, er, eatt, mkey, GAT_E);
        edge_exp_kernel<<<(GAT_E + 255) / 256, 256, 0, stream>>>(dp, eatt, mkey, denom, GAT_E);
        edge_msg_kernel<<<(GAT_E + 3) / 4, 256, 0, stream>>>(sp, dp, eatt, denom, fs, out, GAT_E);
    }
}